// PointCloudShotDescriptor_4252017623541
// MI455X (gfx1250) — compile-verified
//
#include <hip/hip_runtime.h>
#include <math.h>

#define NPTS 4096
#define KNN  16
#define AZB  8
#define ELB  2
#define RDB  2
#define HSB  11
#define DESC (AZB*ELB*RDB*HSB)   // 352
#define EPSF 1e-12f
#define PI_F 3.14159265358979323846f

typedef float v2f __attribute__((ext_vector_type(2)));
typedef float v8f __attribute__((ext_vector_type(8)));

// ---------------------------------------------------------------------------
// Kernel 1: brute-force kNN via V_WMMA_F32_16X16X4_F32 Gram tiles.
// One wave (32 lanes) handles 16 query points; loops over 256 candidate tiles.
// A = candidate tile (M), B = query tile (N), K = 4 (xyz + zero pad).
// d2 = |q|^2 + |c|^2 - 2*dot.  Lanes 0..15 each own one query's top-16 list.
// ---------------------------------------------------------------------------
__global__ __launch_bounds__(32)
void knn_wmma_kernel(const float* __restrict__ vtx, int* __restrict__ nnidx, int N)
{
    const int lane  = threadIdx.x;        // 0..31
    const int l     = lane & 15;
    const int b     = blockIdx.y;
    const int qbase = blockIdx.x * 16;
    const float* v  = vtx + (size_t)b * N * 3;

    // B fragment (4x16 f32): lane<16 -> (K0,K1)=(x,y) of query N=l ; lane>=16 -> (K2,K3)=(z,0)
    v2f bf;
    if (lane < 16) { bf.x = v[(qbase + l)*3 + 0]; bf.y = v[(qbase + l)*3 + 1]; }
    else           { bf.x = v[(qbase + l)*3 + 2]; bf.y = 0.0f; }

    float pb = bf.x*bf.x + bf.y*bf.y;
    float qn = pb + __shfl(pb, l + 16, 32);         // |query|^2 (valid lanes 0..15)

    float bd[KNN]; int bi[KNN];
    #pragma unroll
    for (int j = 0; j < KNN; ++j) { bd[j] = 3.4e38f; bi[j] = 0; }

    const int q = qbase + l;

    for (int t = 0; t < N/16; ++t) {
        const int cbase = t * 16;
        // A fragment (16x4 f32): lane<16 -> (K0,K1) of cand M=l ; lane>=16 -> (K2,K3)
        v2f af;
        if (lane < 16) { af.x = v[(cbase + l)*3 + 0]; af.y = v[(cbase + l)*3 + 1]; }
        else           { af.x = v[(cbase + l)*3 + 2]; af.y = 0.0f; }
        float pa = af.x*af.x + af.y*af.y;
        float cn = pa + __shfl(pa, l + 16, 32);     // lane m(<16): |cand m|^2

        v8f c = {};
        c = __builtin_amdgcn_wmma_f32_16x16x4_f32(
                /*neg_a=*/false, af, /*neg_b=*/false, bf,
                /*c_mod=*/(short)0, c, /*reuse_a=*/false, /*reuse_b=*/false);

        // Gather full column: lane n(<16) gets dots for cands cbase+0..15 vs query n.
        float dall[16];
        #pragma unroll
        for (int r = 0; r < 8; ++r) {
            dall[r]     = c[r];
            dall[r + 8] = __shfl(c[r], l + 16, 32);  // from high-half lanes (active here)
        }

        if (lane < 16) {
            #pragma unroll
            for (int m = 0; m < 16; ++m) {
                const int cand = cbase + m;
                float cnm = __shfl(cn, m, 32);       // source lanes 0..15: active
                float d2  = qn + cnm - 2.0f * dall[m];
                if ((cand != q) && (d2 < bd[KNN-1])) {
                    float vv = d2; int vi = cand;    // branchless sorted insert (asc)
                    #pragma unroll
                    for (int j = 0; j < KNN; ++j) {
                        bool  sw = vv < bd[j];       // strict: ties keep lower index
                        float tf = bd[j]; int ti = bi[j];
                        bd[j] = sw ? vv : bd[j];  bi[j] = sw ? vi : bi[j];
                        vv    = sw ? tf : vv;     vi    = sw ? ti : vi;
                    }
                }
            }
        }
    }

    if (lane < 16) {
        int* o = nnidx + ((size_t)b * N + q) * KNN;
        #pragma unroll
        for (int j = 0; j < KNN; ++j) o[j] = bi[j];
    }
}

// ---------------------------------------------------------------------------
// Jacobi rotation for 3x3 symmetric eigensolve (compile-time indices only).
// ---------------------------------------------------------------------------
template<int p, int q, int r>
__device__ __forceinline__ void jrot(float A[3][3], float V[3][3])
{
    float apq = A[p][q];
    if (fabsf(apq) > 1e-30f) {
        float app = A[p][p], aqq = A[q][q];
        float th  = (aqq - app) / (2.0f * apq);
        float t   = copysignf(1.0f, th) / (fabsf(th) + sqrtf(th*th + 1.0f));
        float cc  = 1.0f / sqrtf(t*t + 1.0f);
        float ss  = t * cc;
        A[p][p] = app - t * apq;
        A[q][q] = aqq + t * apq;
        A[p][q] = 0.0f; A[q][p] = 0.0f;
        float arp = A[r][p], arq = A[r][q];
        A[r][p] = cc*arp - ss*arq; A[p][r] = A[r][p];
        A[r][q] = ss*arp + cc*arq; A[q][r] = A[r][q];
        #pragma unroll
        for (int i = 0; i < 3; ++i) {
            float vip = V[i][p], viq = V[i][q];
            V[i][p] = cc*vip - ss*viq;
            V[i][q] = ss*vip + cc*viq;
        }
    }
}

__device__ __forceinline__ float colget(const float V[3][3], int row, int c)
{
    float a = (row==0)?V[0][0]:((row==1)?V[1][0]:V[2][0]);
    float b = (row==0)?V[0][1]:((row==1)?V[1][1]:V[2][1]);
    float d = (row==0)?V[0][2]:((row==1)?V[1][2]:V[2][2]);
    return (c==0) ? a : ((c==1) ? b : d);
}

// ---------------------------------------------------------------------------
// Kernel 2: per-point LRF (weighted cov -> eigh -> disambiguate -> cross),
// also emits per-point max over rotated neighborhood coords.
// ---------------------------------------------------------------------------
__global__ void lrf_kernel(const float* __restrict__ vtx, const int* __restrict__ nnidx,
                           float* __restrict__ lrf, float* __restrict__ pmax,
                           int N, int total)
{
    int gid = blockIdx.x * blockDim.x + threadIdx.x;
    if (gid >= total) return;
    int b = gid / N, i = gid - b * N;
    const float* v = vtx + (size_t)b * N * 3;
    float qx = v[i*3+0], qy = v[i*3+1], qz = v[i*3+2];

    float nx[KNN], ny[KNN], nz[KNN], dist[KNN];
    const int* idx = nnidx + (size_t)gid * KNN;
    float radius = 0.0f;
    #pragma unroll
    for (int k = 0; k < KNN; ++k) {
        int j = idx[k];
        nx[k] = v[j*3+0] - qx; ny[k] = v[j*3+1] - qy; nz[k] = v[j*3+2] - qz;
        dist[k] = sqrtf(nx[k]*nx[k] + ny[k]*ny[k] + nz[k]*nz[k]);
        radius  = fmaxf(radius, dist[k]);
    }
    float sw = 0.f, c00=0.f, c01=0.f, c02=0.f, c11=0.f, c12=0.f, c22=0.f;
    #pragma unroll
    for (int k = 0; k < KNN; ++k) {
        float w = radius - dist[k];
        sw += w;
        c00 += w*nx[k]*nx[k]; c01 += w*nx[k]*ny[k]; c02 += w*nx[k]*nz[k];
        c11 += w*ny[k]*ny[k]; c12 += w*ny[k]*nz[k]; c22 += w*nz[k]*nz[k];
    }
    float inv = 1.0f / (sw + EPSF);
    float A[3][3] = {{c00*inv, c01*inv, c02*inv},
                     {c01*inv, c11*inv, c12*inv},
                     {c02*inv, c12*inv, c22*inv}};
    float V[3][3] = {{1,0,0},{0,1,0},{0,0,1}};
    #pragma unroll
    for (int s = 0; s < 10; ++s) {
        jrot<0,1,2>(A, V); jrot<0,2,1>(A, V); jrot<1,2,0>(A, V);
    }
    float e0 = A[0][0], e1 = A[1][1], e2 = A[2][2];
    int imin = (e0 <= e1) ? ((e0 <= e2) ? 0 : 2) : ((e1 <= e2) ? 1 : 2);
    int imax = (e0 >= e1) ? ((e0 >= e2) ? 0 : 2) : ((e1 >= e2) ? 1 : 2);
    if (imin == imax) { imin = 0; imax = 2; }

    float z0 = colget(V, 0, imin), z1 = colget(V, 1, imin), z2 = colget(V, 2, imin);
    float x0 = colget(V, 0, imax), x1 = colget(V, 1, imax), x2 = colget(V, 2, imax);

    // Sign disambiguation: keep +axis iff 2*count(dot>=0) >= k
    int posz = 0, posx = 0;
    #pragma unroll
    for (int k = 0; k < KNN; ++k) {
        posz += (nx[k]*z0 + ny[k]*z1 + nz[k]*z2 >= 0.0f);
        posx += (nx[k]*x0 + ny[k]*x1 + nz[k]*x2 >= 0.0f);
    }
    if (2*posz < KNN) { z0 = -z0; z1 = -z1; z2 = -z2; }
    if (2*posx < KNN) { x0 = -x0; x1 = -x1; x2 = -x2; }
    float y0 = z1*x2 - z2*x1;
    float y1 = z2*x0 - z0*x2;
    float y2 = z0*x1 - z1*x0;

    // per-point max over rotated coords (for the global "radius" scalar)
    float m = -3.4e38f;
    #pragma unroll
    for (int k = 0; k < KNN; ++k) {
        float pz = nx[k]*z0 + ny[k]*z1 + nz[k]*z2;
        float py = nx[k]*y0 + ny[k]*y1 + nz[k]*y2;
        float px = nx[k]*x0 + ny[k]*x1 + nz[k]*x2;
        m = fmaxf(m, fmaxf(pz, fmaxf(py, px)));
    }
    pmax[gid] = m;

    float* L = lrf + (size_t)gid * 9;
    L[0]=z0; L[1]=z1; L[2]=z2;
    L[3]=y0; L[4]=y1; L[5]=y2;
    L[6]=x0; L[7]=x1; L[8]=x2;
}

// ---------------------------------------------------------------------------
// Kernel 3: per-cloud max reduce of pmax -> gmax[b]
// ---------------------------------------------------------------------------
__global__ __launch_bounds__(256)
void gmax_kernel(const float* __restrict__ pmax, float* __restrict__ gmax, int N)
{
    __shared__ float s[256];
    int b = blockIdx.x;
    float m = -3.4e38f;
    for (int i = threadIdx.x; i < N; i += 256) m = fmaxf(m, pmax[(size_t)b*N + i]);
    s[threadIdx.x] = m;
    __syncthreads();
    for (int o = 128; o > 0; o >>= 1) {
        if (threadIdx.x < o) s[threadIdx.x] = fmaxf(s[threadIdx.x], s[threadIdx.x + o]);
        __syncthreads();
    }
    if (threadIdx.x == 0) gmax[b] = s[0];
}

// ---------------------------------------------------------------------------
// Kernel 4: SHOT histogram (352 bins) + L2 normalize. One thread per point;
// duplicate-count trick lets us write final normalized values with no RMW.
// ---------------------------------------------------------------------------
__device__ __forceinline__ int binf(float x, int nb)
{
    float f = floorf(x);
    f = fminf(fmaxf(f, 0.0f), (float)(nb - 1));
    return (int)f;
}

__global__ void descr_kernel(const float* __restrict__ vtx, const int* __restrict__ nnidx,
                             const float* __restrict__ lrf, const float* __restrict__ gmax,
                             float* __restrict__ out, int N, int total)
{
    int gid = blockIdx.x * blockDim.x + threadIdx.x;
    if (gid >= total) return;
    int b = gid / N, i = gid - b * N;
    const float* v = vtx + (size_t)b * N * 3;
    const float* L = lrf + (size_t)gid * 9;
    float z0=L[0], z1=L[1], z2=L[2];
    float y0=L[3], y1=L[4], y2=L[5];
    float x0=L[6], x1=L[7], x2=L[8];
    float qx = v[i*3+0], qy = v[i*3+1], qz = v[i*3+2];
    float radius = gmax[b];
    const int* idx = nnidx + (size_t)gid * KNN;
    float* o = out + (size_t)gid * DESC;

    #pragma unroll 4
    for (int t = 0; t < DESC; ++t) o[t] = 0.0f;

    int f[KNN];
    #pragma unroll
    for (int k = 0; k < KNN; ++k) {
        int j = idx[k];
        float dx = v[j*3+0] - qx, dy = v[j*3+1] - qy, dz = v[j*3+2] - qz;
        float pz = dx*z0 + dy*z1 + dz*z2;
        float py = dx*y0 + dy*y1 + dz*y2;
        float px = dx*x0 + dy*x1 + dz*x2;
        float r  = sqrtf(pz*pz + py*py + px*px);
        float az = atan2f(py, px);
        int azb = binf((az + PI_F) / (2.0f * PI_F) * (float)AZB, AZB);
        float sel = pz / (r + EPSF);
        int elb = binf((sel + 1.0f) * 0.5f * (float)ELB, ELB);
        int rdb = binf(r / (radius + EPSF) * (float)RDB, RDB);
        const float* Lj = lrf + ((size_t)b * N + j) * 9;
        float cosn = Lj[0]*z0 + Lj[1]*z1 + Lj[2]*z2;   // neighbor normal . own normal
        int hsb = binf((cosn + 1.0f) * 0.5f * (float)HSB, HSB);
        f[k] = ((azb*ELB + elb)*RDB + rdb)*HSB + hsb;
    }

    // ||descr||^2 = sum over entries of (count of its bin)
    float n2 = 0.0f;
    float cnt[KNN];
    #pragma unroll
    for (int e = 0; e < KNN; ++e) {
        int c = 0;
        #pragma unroll
        for (int e2 = 0; e2 < KNN; ++e2) c += (f[e2] == f[e]);
        cnt[e] = (float)c;
        n2 += (float)c;
    }
    float invn = 1.0f / (sqrtf(n2) + EPSF);
    #pragma unroll
    for (int e = 0; e < KNN; ++e) o[f[e]] = cnt[e] * invn;  // idempotent per bin
}

// ---------------------------------------------------------------------------
extern "C" void kernel_launch(void* const* d_in, const int* in_sizes, int n_in,
                              void* d_out, int out_size, void* d_ws, size_t ws_size,
                              hipStream_t stream)
{
    const float* vtx = (const float*)d_in[0];
    const int N = NPTS;
    const int B = (in_sizes[0] / 3) / N;   // 4
    const int total = B * N;

    char* ws = (char*)d_ws;
    size_t off = 0;
    int*   nnidx = (int*)(ws + off);   off += (size_t)total * KNN * sizeof(int);   // 1 MB
    float* lrf   = (float*)(ws + off); off += (size_t)total * 9   * sizeof(float); // 576 KB
    float* pmax  = (float*)(ws + off); off += (size_t)total       * sizeof(float); // 64 KB
    float* gmax  = (float*)(ws + off);

    dim3 g1(N / 16, B);
    knn_wmma_kernel<<<g1, 32, 0, stream>>>(vtx, nnidx, N);
    lrf_kernel<<<(total + 127) / 128, 128, 0, stream>>>(vtx, nnidx, lrf, pmax, N, total);
    gmax_kernel<<<B, 256, 0, stream>>>(pmax, gmax, N);
    descr_kernel<<<(total + 63) / 64, 64, 0, stream>>>(vtx, nnidx, lrf, gmax,
                                                       (float*)d_out, N, total);
}